// ObjectCondensationLoss_31138512896539
// MI455X (gfx1250) — compile-verified
//
#include <hip/hip_runtime.h>
#include <hip/hip_bf16.h>

// Problem constants (from reference setup_inputs)
#define BB 4
#define NN 4096
#define DD 8
#define KK 64

// Workspace layout (in 32-bit words). Total ~131 KB.
#define WS_POS    0        // int  [4]   CP count per batch (also compaction counter)
#define WS_S1     4        // f32  [4]   sum of y*softplus(-beta)
#define WS_S0     8        // f32  [4]   sum of (1-y)*softplus(beta)
#define WS_REP    12       // f32  [4]   repulsion pair sum
#define WS_FCP    16       // int  [4*64] first CP index per (b,slice), init INT_MAX
#define WS_CNT    272      // int  [4*64] segment counts
#define WS_SEG    528      // f32  [4*64] segment d2 sums
#define WS_CPIDX  784      // int  [4*4096] compacted CP point indices
#define WS_SQCP   17168    // f32  [4*4096] |e|^2 of compacted CP points
#define WS_WORDS  33552

typedef __attribute__((ext_vector_type(2)))  float v2f;
typedef __attribute__((ext_vector_type(8)))  float v8f;

__device__ __forceinline__ float softplusf(float x) {
  // numerically stable softplus
  return (x > 0.0f) ? x + log1pf(__expf(-x)) : log1pf(__expf(x));
}

__device__ __forceinline__ float waveReduceAdd(float v) {
  #pragma unroll
  for (int off = 16; off > 0; off >>= 1) v += __shfl_xor(v, off, 32);
  return v;
}

// ---------------------------------------------------------------------------
// ws init: must run every launch (graph replay, ws is not re-poisoned)
__global__ void oc_init_ws(int* ws) {
  int t = blockIdx.x * blockDim.x + threadIdx.x;
  if (t < WS_CPIDX) {
    ws[t] = (t >= WS_FCP && t < WS_CNT) ? 0x7fffffff : 0;  // firstcp=+inf, rest 0
  }
}

// ---------------------------------------------------------------------------
// pass1: BCE partial sums, per-slice count + first-CP (LDS histogram),
//        CP compaction (global atomic slot), |e|^2 of compacted points.
// 64 blocks x 256 threads; each block lies entirely inside one batch.
__global__ void oc_pass1(const float* __restrict__ beta,
                         const float* __restrict__ embed,
                         const int*   __restrict__ slice_id,
                         const int*   __restrict__ is_cp,
                         int* pos, float* S1, float* S0,
                         int* firstcp, int* counts,
                         int* cpIdx, float* sqCp) {
  __shared__ int   l_cnt[KK];
  __shared__ int   l_fcp[KK];
  __shared__ float l_s1[8], l_s0[8];

  const int t    = threadIdx.x;
  const int gid  = blockIdx.x * 256 + t;
  const int b    = gid >> 12;       // /4096, uniform per block
  const int n    = gid & (NN - 1);
  const int lane = t & 31;
  const int wv   = t >> 5;

  if (t < KK) { l_cnt[t] = 0; l_fcp[t] = 0x7fffffff; }
  __syncthreads();

  const float bt = beta[gid];
  const int   cp = is_cp[gid];
  const int   sd = slice_id[gid];

  float s1 = 0.0f, s0 = 0.0f;
  if (cp == 1) s1 = softplusf(-bt); else s0 = softplusf(bt);

  const float* e = embed + (size_t)gid * DD;
  float sq = 0.0f;
  #pragma unroll
  for (int d = 0; d < DD; ++d) sq += e[d] * e[d];

  atomicAdd(&l_cnt[sd], 1);
  if (cp == 1) {
    atomicMin(&l_fcp[sd], n);
    int slot = atomicAdd(&pos[b], 1);     // compaction
    cpIdx[b * NN + slot] = n;
    sqCp [b * NN + slot] = sq;
  }

  // block-reduce the two BCE partial sums -> one global atomic each
  float r1 = waveReduceAdd(s1);
  float r0 = waveReduceAdd(s0);
  if (lane == 0) { l_s1[wv] = r1; l_s0[wv] = r0; }
  __syncthreads();
  if (t == 0) {
    float a1 = 0.0f, a0 = 0.0f;
    #pragma unroll
    for (int i = 0; i < 8; ++i) { a1 += l_s1[i]; a0 += l_s0[i]; }
    atomicAdd(&S1[b], a1);
    atomicAdd(&S0[b], a0);
  }
  if (t < KK) {
    if (l_cnt[t] > 0)             atomicAdd(&counts [b * KK + t], l_cnt[t]);
    if (l_fcp[t] != 0x7fffffff)   atomicMin(&firstcp[b * KK + t], l_fcp[t]);
  }
}

// ---------------------------------------------------------------------------
// pass2: attraction distances to first CP of own slice, LDS-binned segment sums
__global__ void oc_pass2(const float* __restrict__ embed,
                         const int*   __restrict__ slice_id,
                         const int*   __restrict__ firstcp,
                         float* segsum) {
  __shared__ float l_ss[KK];
  const int t   = threadIdx.x;
  const int gid = blockIdx.x * 256 + t;
  const int b   = gid >> 12;

  if (t < KK) l_ss[t] = 0.0f;
  __syncthreads();

  const int sd  = slice_id[gid];
  int fc = firstcp[b * KK + sd];
  fc = (fc > NN - 1) ? NN - 1 : fc;     // jnp.clip(first_cp, 0, N-1)

  const float* e = embed + (size_t)gid * DD;
  const float* c = embed + ((size_t)(b << 12) + fc) * DD;
  float d2 = 0.0f;
  #pragma unroll
  for (int d = 0; d < DD; ++d) { float df = e[d] - c[d]; d2 += df * df; }

  atomicAdd(&l_ss[sd], d2);
  __syncthreads();
  if (t < KK) atomicAdd(&segsum[b * KK + t], l_ss[t]);
}

// ---------------------------------------------------------------------------
// rep_tiles: CP x CP pairwise exp(-d2) sum via WMMA GEMM on compacted points.
// grid (32, 256, B), block 256 (= 8 waves); wave w handles tile
// (blockIdx.x*8 + w, blockIdx.y). Tiles beyond ceil(pos/16) exit (uniform
// per wave at runtime). Epilogue is fully branchless: pad rows/cols have
// clamped, finite d2 (exp <= 1) and are zeroed by the 0/1 mask wi*wj.
__global__ void oc_rep_tiles(const float* __restrict__ embed,
                             const int*   __restrict__ posArr,
                             const int*   __restrict__ cpIdx,
                             const float* __restrict__ sqCp,
                             float* repsum) {
  const int b     = blockIdx.z;
  const int lane  = threadIdx.x & 31;
  const int iTile = blockIdx.x * 8 + (threadIdx.x >> 5);
  const int jTile = blockIdx.y;

  const int pos = posArr[b];
  const int nT  = (pos + 15) >> 4;
  if (iTile >= nT || jTile >= nT) return;   // wave-uniform at runtime

  const int m    = lane & 15;
  const int kh   = lane >> 4;
  const int last = pos - 1;                 // pos >= 1 here
  const int ri   = iTile * 16 + m;
  const int rj   = jTile * 16 + m;
  const int ci   = (ri < pos) ? ri : last;  // clamped compact indices
  const int cj   = (rj < pos) ? rj : last;
  const int gi   = cpIdx[b * NN + ci];
  const int gj   = cpIdx[b * NN + cj];

  // f32 A 16x4 / B 4x16 layout: per-lane float2 at row*8 + 2*(lane>>4);
  // +4 element offset for the second K-step. A and B indexing is identical
  // (symmetric GEMM), with i-tile rows vs j-tile rows.
  const float* Ei = embed + ((size_t)b * NN + gi) * DD + 2 * kh;
  const float* Ej = embed + ((size_t)b * NN + gj) * DD + 2 * kh;
  v2f a0 = *(const v2f*)(Ei);
  v2f b0 = *(const v2f*)(Ej);
  v2f a1 = *(const v2f*)(Ei + 4);
  v2f b1 = *(const v2f*)(Ej + 4);

  v8f acc = {};
  acc = __builtin_amdgcn_wmma_f32_16x16x4_f32(false, a0, false, b0,
                                              (short)0, acc, false, false);
  acc = __builtin_amdgcn_wmma_f32_16x16x4_f32(false, a1, false, b1,
                                              (short)0, acc, false, false);

  // |e|^2 per row of the i-tile lives in this wave already: lane m holds
  // row m (lanes 0-15). Broadcast row r via shuffle instead of 8 loads.
  const float sqi_lane = sqCp[b * NN + ci];
  const float sqj      = sqCp[b * NN + cj];
  const float wj       = (rj < pos) ? 1.0f : 0.0f;

  // C/D layout: lane L, elem v -> G[iTile*16 + (L>>4)*8 + v][jTile*16 + (L&15)]
  float s = 0.0f;
  #pragma unroll
  for (int v = 0; v < 8; ++v) {
    const int r  = kh * 8 + v;                    // row within tile (0..15)
    const float sqi = __shfl(sqi_lane, r, 32);    // from lane r (holds row r)
    const float wi  = ((iTile * 16 + r) < pos) ? 1.0f : 0.0f;
    float d2 = fmaxf(sqi + sqj - 2.0f * acc[v], 0.0f);
    s += (wi * wj) * __expf(-d2);
  }

  s = waveReduceAdd(s);
  if (lane == 0) atomicAdd(&repsum[b], s);
}

// ---------------------------------------------------------------------------
// finalize: combine terms exactly as reference; single thread (trivial work)
__global__ void oc_finalize(const int* pos, const float* S1, const float* S0,
                            const int* firstcp, const int* counts,
                            const float* segsum, const float* repsum,
                            float* out) {
  float tot = 0.0f, cnt = 0.0f;
  for (int b = 0; b < BB; ++b) {
    const float p  = (float)pos[b];
    const float ng = (float)NN - p;
    const bool valid = (p >= 1.0f) && (ng >= 1.0f);
    const float pw  = ng / (p + 1e-6f);
    const float bce = (pw * S1[b] + S0[b]) * (1.0f / (float)NN);
    float attr = 0.0f;
    for (int k = 0; k < KK; ++k) {
      const float c  = (float)counts[b * KK + k];
      const bool has = firstcp[b * KK + k] < NN;
      if (c > 0.0f && has) attr += segsum[b * KK + k] / fmaxf(c, 1.0f);
    }
    const float rep = (p > 1.0f) ? repsum[b] / fmaxf(p * p, 1.0f) : 0.0f;
    if (valid) { tot += bce + attr + rep; cnt += 1.0f; }
  }
  out[0] = (cnt > 0.0f) ? tot / fmaxf(cnt, 1.0f) : 0.0f;
}

// ---------------------------------------------------------------------------
extern "C" void kernel_launch(void* const* d_in, const int* in_sizes, int n_in,
                              void* d_out, int out_size, void* d_ws, size_t ws_size,
                              hipStream_t stream) {
  const float* beta     = (const float*)d_in[0];   // [B,N,1] f32
  const float* embed    = (const float*)d_in[1];   // [B,N,D] f32
  const int*   slice_id = (const int*)  d_in[2];   // [B,N] i32
  const int*   is_cp    = (const int*)  d_in[3];   // [B,N] i32
  float* out = (float*)d_out;

  int*   wsI = (int*)d_ws;
  float* wsF = (float*)d_ws;
  (void)in_sizes; (void)n_in; (void)out_size; (void)ws_size;

  oc_init_ws<<<1, 1024, 0, stream>>>(wsI);

  oc_pass1<<<(BB * NN) / 256, 256, 0, stream>>>(
      beta, embed, slice_id, is_cp,
      wsI + WS_POS, wsF + WS_S1, wsF + WS_S0,
      wsI + WS_FCP, wsI + WS_CNT,
      wsI + WS_CPIDX, wsF + WS_SQCP);

  oc_pass2<<<(BB * NN) / 256, 256, 0, stream>>>(
      embed, slice_id, wsI + WS_FCP, wsF + WS_SEG);

  dim3 grid(32, 256, BB);
  oc_rep_tiles<<<grid, 256, 0, stream>>>(
      embed, wsI + WS_POS, wsI + WS_CPIDX, wsF + WS_SQCP, wsF + WS_REP);

  oc_finalize<<<1, 1, 0, stream>>>(
      wsI + WS_POS, wsF + WS_S1, wsF + WS_S0,
      wsI + WS_FCP, wsI + WS_CNT, wsF + WS_SEG, wsF + WS_REP, out);
}